// MVFuse_3264175145351
// MI455X (gfx1250) — compile-verified
//
#include <hip/hip_runtime.h>
#include <math.h>

// Problem constants (from reference)
#define NB    2
#define NPTS  160000
#define C0    64
#define CIN   192
#define COUT  64
#define NCLS  20
#define BH    512
#define BW    512
#define RH    64
#define RW    2048

typedef __attribute__((ext_vector_type(2))) float v2f;
typedef __attribute__((ext_vector_type(8))) float v8f;

__device__ __forceinline__ v8f vzero8() {
    v8f z = {0.f, 0.f, 0.f, 0.f, 0.f, 0.f, 0.f, 0.f};
    return z;
}

// clipped + masked fetch (matches reference zero-padding semantics)
__device__ __forceinline__ float fetch_c(const float* __restrict__ base,
                                         int x, int y, int W, int H) {
    bool ok = (x >= 0) & (x < W) & (y >= 0) & (y < H);
    int xc = min(max(x, 0), W - 1);
    int yc = min(max(y, 0), H - 1);
    float v = base[(size_t)yc * W + xc];
    return ok ? v : 0.0f;
}

__global__ __launch_bounds__(32) void mvfuse_kernel(
    const float* __restrict__ pf,        // [B,64,N]
    const float* __restrict__ bev_sem,   // [B,64,512,512]
    const float* __restrict__ bev_ins,
    const float* __restrict__ coord_bev, // [B,N,2]
    const float* __restrict__ rv_sem,    // [B,64,64,2048]
    const float* __restrict__ rv_ins,
    const float* __restrict__ coord_rv,
    const float* __restrict__ w_sem,     // [64,192]
    const float* __restrict__ b_sem,     // [64]
    const float* __restrict__ w_ins,
    const float* __restrict__ b_ins,
    const float* __restrict__ w_pred,    // [20,64]
    const float* __restrict__ b_pred,    // [20]
    const float* __restrict__ w_off,     // [3,64]
    const float* __restrict__ b_off,     // [3]
    const float* __restrict__ w_hmap,    // [1,64]
    const float* __restrict__ b_hmap,    // [1]
    float* __restrict__ out_feat,        // [B,64,N]
    float* __restrict__ out_pred,        // [B,20,N]
    float* __restrict__ out_off,         // [B,N,3]
    float* __restrict__ out_hmap)        // [B,N]
{
    __shared__ float X[CIN * 16];   // activation tile: rows 0-63 pf, 64-127 bev, 128-191 rv
    __shared__ float F[COUT * 16];  // feat tile staged for head GEMMs (B layout)
    __shared__ int   sx0[2][16], sy0[2][16];
    __shared__ float swx[2][16], swy[2][16];

    const int lane = threadIdx.x;
    const int n0   = blockIdx.x * 16;
    const int b    = blockIdx.y;
    const int half = lane >> 4;     // 0: lanes 0-15, 1: lanes 16-31
    const int l15  = lane & 15;

    // ---- per-point bilinear params: lanes 0-15 -> bev, lanes 16-31 -> rv
    {
        const float* cp = half ? coord_rv : coord_bev;
        const float Wm  = half ? (float)RW : (float)BW;
        const float Hm  = half ? (float)RH : (float)BH;
        float2 xy = ((const float2*)cp)[(size_t)b * NPTS + n0 + l15];
        float gx = xy.x * Wm - 0.5f;
        float gy = xy.y * Hm - 0.5f;
        float fx = floorf(gx), fy = floorf(gy);
        sx0[half][l15] = (int)fx;
        sy0[half][l15] = (int)fy;
        swx[half][l15] = gx - fx;
        swy[half][l15] = gy - fy;
    }

    // ---- point-feature rows 0..63 (only filled once; shared by sem & ins)
    for (int i = 0; i < 32; ++i) {
        int idx = i * 32 + lane;
        int ch = idx >> 4, p = idx & 15;
        X[ch * 16 + p] = pf[((size_t)b * C0 + ch) * NPTS + n0 + p];
    }
    __syncthreads();

    // ---- bilinear sampler: fills 64 rows of X starting at rowbase
    auto sample = [&](const float* __restrict__ fmap, int m, int Wm, int Hm, int rowbase) {
        const size_t mb = (size_t)b * C0 * (size_t)Hm * Wm;
        for (int i = 0; i < 32; ++i) {
            int idx = i * 32 + lane;
            int ch = idx >> 4, p = idx & 15;
            int x0 = sx0[m][p], y0 = sy0[m][p];
            float wx = swx[m][p], wy = swy[m][p];
            const float* base = fmap + mb + (size_t)ch * Hm * Wm;
            float v00 = fetch_c(base, x0,     y0,     Wm, Hm);
            float v10 = fetch_c(base, x0 + 1, y0,     Wm, Hm);
            float v01 = fetch_c(base, x0,     y0 + 1, Wm, Hm);
            float v11 = fetch_c(base, x0 + 1, y0 + 1, Wm, Hm);
            float ix = 1.0f - wx, iy = 1.0f - wy;
            X[(rowbase + ch) * 16 + p] = (v00 * ix + v10 * wx) * iy
                                       + (v01 * ix + v11 * wx) * wy;
        }
    };

    // ---- fuse GEMM: D[64,16] = W[64,192] * X[192,16] via v_wmma_f32_16x16x4_f32
    auto fuse_gemm = [&](const float* __restrict__ Wt, v8f acc[4]) {
        #pragma unroll 4
        for (int ks = 0; ks < 48; ++ks) {
            int k0 = ks * 4 + half * 2;
            v2f bt;
            bt.x = X[k0 * 16 + l15];
            bt.y = X[(k0 + 1) * 16 + l15];
            #pragma unroll
            for (int mt = 0; mt < 4; ++mt) {
                const float* wr = Wt + (size_t)(mt * 16 + l15) * CIN + k0;
                v2f at;
                at.x = wr[0];
                at.y = wr[1];
                acc[mt] = __builtin_amdgcn_wmma_f32_16x16x4_f32(
                    false, at, false, bt, (short)0, acc[mt], false, false);
            }
        }
    };

    // ---- bias + relu; stage into F (B layout) and optionally store feat
    auto fuse_epi = [&](v8f acc[4], const float* __restrict__ bias,
                        float* __restrict__ outp) {
        #pragma unroll
        for (int mt = 0; mt < 4; ++mt) {
            #pragma unroll
            for (int r = 0; r < 8; ++r) {
                int row = mt * 16 + half * 8 + r;   // C/D layout -> row index
                float v = fmaxf(acc[mt][r] + bias[row], 0.0f);
                F[row * 16 + l15] = v;
                if (outp) outp[((size_t)b * COUT + row) * NPTS + n0 + l15] = v;
            }
        }
    };

    // ---- head GEMM over K=64 from F; weights via callback (zero-padded rows)
    auto head_gemm = [&](auto getw, int ntiles, v8f* acc) {
        #pragma unroll 4
        for (int ks = 0; ks < 16; ++ks) {
            int k0 = ks * 4 + half * 2;
            v2f bt;
            bt.x = F[k0 * 16 + l15];
            bt.y = F[(k0 + 1) * 16 + l15];
            for (int mt = 0; mt < ntiles; ++mt) {
                int m = mt * 16 + l15;
                v2f at;
                at.x = getw(m, k0);
                at.y = getw(m, k0 + 1);
                acc[mt] = __builtin_amdgcn_wmma_f32_16x16x4_f32(
                    false, at, false, bt, (short)0, acc[mt], false, false);
            }
        }
    };

    // ================= SEM branch =================
    sample(bev_sem, 0, BW, BH, 64);
    sample(rv_sem, 1, RW, RH, 128);
    __syncthreads();

    v8f acc[4] = {vzero8(), vzero8(), vzero8(), vzero8()};
    fuse_gemm(w_sem, acc);
    fuse_epi(acc, b_sem, out_feat);
    __syncthreads();

    // pred_sem: [20,64] padded to 2 M-tiles of 16
    {
        v8f pacc[2] = {vzero8(), vzero8()};
        head_gemm([&](int m, int k) -> float {
            return (m < NCLS) ? w_pred[(size_t)m * COUT + k] : 0.0f;
        }, 2, pacc);
        #pragma unroll
        for (int mt = 0; mt < 2; ++mt) {
            #pragma unroll
            for (int r = 0; r < 8; ++r) {
                int row = mt * 16 + half * 8 + r;
                if (row < NCLS)
                    out_pred[((size_t)b * NCLS + row) * NPTS + n0 + l15] =
                        pacc[mt][r] + b_pred[row];
            }
        }
    }
    __syncthreads();

    // ================= INS branch =================
    sample(bev_ins, 0, BW, BH, 64);
    sample(rv_ins, 1, RW, RH, 128);
    __syncthreads();

    acc[0] = vzero8(); acc[1] = vzero8(); acc[2] = vzero8(); acc[3] = vzero8();
    fuse_gemm(w_ins, acc);
    fuse_epi(acc, b_ins, nullptr);   // feat_ins is internal only
    __syncthreads();

    // offset (3 rows) + heatmap (row 3), one padded 16-row M-tile
    {
        v8f oacc[1] = {vzero8()};
        head_gemm([&](int m, int k) -> float {
            if (m < 3) return w_off[(size_t)m * COUT + k];
            if (m == 3) return w_hmap[k];
            return 0.0f;
        }, 1, oacc);
        if (half == 0) {   // rows 0-7 live in lanes 0-15
            #pragma unroll
            for (int r = 0; r < 3; ++r)
                out_off[((size_t)b * NPTS + n0 + l15) * 3 + r] = oacc[0][r] + b_off[r];
            float h = oacc[0][3] + b_hmap[0];
            out_hmap[(size_t)b * NPTS + n0 + l15] = 1.0f / (1.0f + expf(-h));
        }
    }
}

extern "C" void kernel_launch(void* const* d_in, const int* in_sizes, int n_in,
                              void* d_out, int out_size, void* d_ws, size_t ws_size,
                              hipStream_t stream) {
    (void)in_sizes; (void)n_in; (void)out_size; (void)d_ws; (void)ws_size;

    const float* pf        = (const float*)d_in[0];   // point_feat_tmp [B,64,N,1]
    const float* bev_sem   = (const float*)d_in[1];
    const float* bev_ins   = (const float*)d_in[2];
    const float* coord_bev = (const float*)d_in[3];   // [B,N,1,2]
    const float* rv_sem    = (const float*)d_in[4];
    const float* rv_ins    = (const float*)d_in[5];
    const float* coord_rv  = (const float*)d_in[6];
    const float* w_sem     = (const float*)d_in[7];
    const float* b_sem     = (const float*)d_in[8];
    const float* w_ins     = (const float*)d_in[9];
    const float* b_ins     = (const float*)d_in[10];
    const float* w_pred    = (const float*)d_in[11];
    const float* b_pred    = (const float*)d_in[12];
    const float* w_off     = (const float*)d_in[13];
    const float* b_off     = (const float*)d_in[14];
    const float* w_hmap    = (const float*)d_in[15];
    const float* b_hmap    = (const float*)d_in[16];

    float* out = (float*)d_out;
    constexpr size_t OFF_FEAT = 0;
    constexpr size_t OFF_PRED = (size_t)NB * COUT * NPTS;               // 20,480,000
    constexpr size_t OFF_OFF  = OFF_PRED + (size_t)NB * NCLS * NPTS;    // 26,880,000
    constexpr size_t OFF_HMAP = OFF_OFF + (size_t)NB * NPTS * 3;        // 27,840,000

    dim3 grid(NPTS / 16, NB);
    mvfuse_kernel<<<grid, 32, 0, stream>>>(
        pf, bev_sem, bev_ins, coord_bev, rv_sem, rv_ins, coord_rv,
        w_sem, b_sem, w_ins, b_ins, w_pred, b_pred, w_off, b_off, w_hmap, b_hmap,
        out + OFF_FEAT, out + OFF_PRED, out + OFF_OFF, out + OFF_HMAP);
}